// GatLinear_rel_44727789421068
// MI455X (gfx1250) — compile-verified
//
#include <hip/hip_runtime.h>
#include <hip/hip_bf16.h>

#define Bsz 32
#define Nsz 2048
#define Dsz 512
#define NEGC 1.0e30f
#define NSPLIT 8

typedef __attribute__((ext_vector_type(2))) float v2f;
typedef __attribute__((ext_vector_type(4))) float v4f;
typedef __attribute__((ext_vector_type(8))) float v8f;

// ---------------------------------------------------------------------------
// Kernel 1: kdot[b][n] = dot(K[b,n,:], wk) using V_WMMA_F32_16X16X4_F32.
// Per wave: A = 16 rows x 4 cols of K (ISA layout: lanes 0-15 carry K-cols
// {c,c+1}, lanes 16-31 carry {c+4,c+5}; c = 8j, then 8j+2); B broadcasts the matching wk chunk
// into all 16 N-columns, so every column of the f32 accumulator holds the
// exact row-dot. One float4 load per lane feeds two WMMA steps.
// ---------------------------------------------------------------------------
__global__ __launch_bounds__(256) void kdot_wmma_kernel(
    const float* __restrict__ K, const float* __restrict__ W,
    float* __restrict__ kdot)
{
    __shared__ __align__(16) float lds_wk[Dsz];
    const int b     = blockIdx.x;
    const int chunk = blockIdx.y;          // 16 chunks of 128 rows
    const int t     = threadIdx.x;
    const int lane  = t & 31;
    const int wave  = t >> 5;              // 8 waves, 16 rows each
    const int hi    = lane >> 4;           // 0: lanes 0-15, 1: lanes 16-31
    const int m     = lane & 15;           // row within the 16-row tile

    // wk = W[D .. 2D)
    lds_wk[t]       = W[Dsz + t];
    lds_wk[t + 256] = W[Dsz + t + 256];
    __syncthreads();

    const int row0 = chunk * 128 + wave * 16;
    const float* Krow = K + ((size_t)b * Nsz + (size_t)(row0 + m)) * Dsz;

    v8f acc = {0.f, 0.f, 0.f, 0.f, 0.f, 0.f, 0.f, 0.f};
    #pragma unroll 4
    for (int j = 0; j < Dsz / 8; ++j) {
        const int c = 8 * j + hi * 4;
        v4f a4 = *(const v4f*)(Krow + c);
        v4f b4 = *(const v4f*)(lds_wk + c);
        v2f a0 = {a4.x, a4.y};
        v2f b0 = {b4.x, b4.y};
        acc = __builtin_amdgcn_wmma_f32_16x16x4_f32(false, a0, false, b0,
                                                    (short)0, acc, false, false);
        v2f a1 = {a4.z, a4.w};
        v2f b1 = {b4.z, b4.w};
        acc = __builtin_amdgcn_wmma_f32_16x16x4_f32(false, a1, false, b1,
                                                    (short)0, acc, false, false);
    }

    // C/D layout: VGPR r -> M=r (lanes 0-15), M=8+r (lanes 16-31); columns
    // are identical, so lanes 0 and 16 spill the 16 dots.
    if (m == 0) {
        float* outp = kdot + (size_t)b * Nsz + row0 + hi * 8;
        #pragma unroll
        for (int r = 0; r < 8; ++r) outp[r] = acc[r];
    }
}

// ---------------------------------------------------------------------------
// Kernel 2: per-batch scalars + mask + softmax over N=2048 -> attn_weight.
// ---------------------------------------------------------------------------
__global__ __launch_bounds__(256) void softmax_kernel(
    const float* __restrict__ Q, const int* __restrict__ adj,
    const long long* __restrict__ s_mask, const float* __restrict__ W,
    const float* __restrict__ bias, const float* __restrict__ rel_table,
    const float* __restrict__ kdot, float* __restrict__ out)
{
    __shared__ float lds_a[Nsz];
    __shared__ float lds_red[256];
    __shared__ float lds_s[3];
    const int b = blockIdx.x;
    const int t = threadIdx.x;

    // qdot = Q[b]·wq ; r0 = rel_table[0]·wr ; r1 = rel_table[1]·wr
    const float* Qb = Q + (size_t)b * Dsz;
    const float wr0 = W[2 * Dsz + t], wr1 = W[2 * Dsz + t + 256];
    float pq = Qb[t] * W[t] + Qb[t + 256] * W[t + 256];
    float p0 = rel_table[t] * wr0 + rel_table[t + 256] * wr1;
    float p1 = rel_table[Dsz + t] * wr0 + rel_table[Dsz + t + 256] * wr1;

    for (int which = 0; which < 3; ++which) {
        lds_red[t] = (which == 0) ? pq : (which == 1) ? p0 : p1;
        __syncthreads();
        for (int s = 128; s > 0; s >>= 1) {
            if (t < s) lds_red[t] += lds_red[t + s];
            __syncthreads();
        }
        if (t == 0) lds_s[which] = lds_red[0];
        __syncthreads();
    }

    const float qdot = lds_s[0], rd0 = lds_s[1], rd1 = lds_s[2];
    const float bb = bias[0];
    const int* adjb = adj + b * Nsz;
    const long long* smb = s_mask + (size_t)b * Nsz;
    const float* kd = kdot + (size_t)b * Nsz;

    float lmax = -__builtin_inff();
    #pragma unroll
    for (int i = 0; i < Nsz / 256; ++i) {
        const int n = t + 256 * i;
        float a = kd[n] + qdot + (smb[n] != 0 ? rd1 : rd0) + bb;
        a -= (1.0f - (float)adjb[n]) * NEGC;
        lds_a[n] = a;
        lmax = fmaxf(lmax, a);
    }
    lds_red[t] = lmax; __syncthreads();
    for (int s = 128; s > 0; s >>= 1) {
        if (t < s) lds_red[t] = fmaxf(lds_red[t], lds_red[t + s]);
        __syncthreads();
    }
    const float amax = lds_red[0];
    __syncthreads();

    float lsum = 0.f;
    #pragma unroll
    for (int i = 0; i < Nsz / 256; ++i) {
        const int n = t + 256 * i;
        float e = expf(lds_a[n] - amax);
        lds_a[n] = e;
        lsum += e;
    }
    lds_red[t] = lsum; __syncthreads();
    for (int s = 128; s > 0; s >>= 1) {
        if (t < s) lds_red[t] += lds_red[t + s];
        __syncthreads();
    }
    const float inv = 1.0f / lds_red[0];
    __syncthreads();

    #pragma unroll
    for (int i = 0; i < Nsz / 256; ++i) {
        const int n = t + 256 * i;
        out[(size_t)b * Nsz + n] = lds_a[n] * inv;
    }
}

// ---------------------------------------------------------------------------
// Kernel 3: partial attn_sum over an N-slice (deterministic, no atomics).
// grid (B, D/256, NSPLIT); fully coalesced 1KB rows of V per iteration.
// ---------------------------------------------------------------------------
__global__ __launch_bounds__(256) void wsum_partial_kernel(
    const float* __restrict__ V, const float* __restrict__ w,
    float* __restrict__ part)
{
    __shared__ float lw[Nsz / NSPLIT];
    const int b = blockIdx.x, dc = blockIdx.y, nc = blockIdx.z;
    const int t = threadIdx.x;
    lw[t] = w[(size_t)b * Nsz + nc * (Nsz / NSPLIT) + t];
    __syncthreads();

    const int d = dc * 256 + t;
    const float* Vp = V + ((size_t)b * Nsz + (size_t)nc * (Nsz / NSPLIT)) * Dsz + d;
    float acc = 0.f;
    #pragma unroll 8
    for (int i = 0; i < Nsz / NSPLIT; ++i)
        acc = fmaf(lw[i], Vp[(size_t)i * Dsz], acc);

    part[((size_t)b * NSPLIT + nc) * Dsz + d] = acc;
}

// ---------------------------------------------------------------------------
// Kernel 4: reduce NSPLIT partials -> attn_sum (second output region).
// ---------------------------------------------------------------------------
__global__ __launch_bounds__(256) void wsum_reduce_kernel(
    const float* __restrict__ part, float* __restrict__ out)
{
    const int idx = blockIdx.x * 256 + threadIdx.x;   // 0 .. B*D-1
    const int b = idx / Dsz, d = idx - b * Dsz;
    float s = 0.f;
    #pragma unroll
    for (int k = 0; k < NSPLIT; ++k)
        s += part[((size_t)b * NSPLIT + k) * Dsz + d];
    out[Bsz * Nsz + idx] = s;
}

extern "C" void kernel_launch(void* const* d_in, const int* in_sizes, int n_in,
                              void* d_out, int out_size, void* d_ws, size_t ws_size,
                              hipStream_t stream) {
    const float*     Q     = (const float*)d_in[0];
    const float*     K     = (const float*)d_in[1];
    const float*     V     = (const float*)d_in[2];
    const int*       adj   = (const int*)d_in[3];
    const long long* smask = (const long long*)d_in[4];
    const float*     W     = (const float*)d_in[5];
    const float*     bias  = (const float*)d_in[6];
    const float*     rel   = (const float*)d_in[7];
    float* out  = (float*)d_out;
    float* kdot = (float*)d_ws;                 // B*N floats   (256 KB)
    float* part = kdot + (size_t)Bsz * Nsz;     // B*NSPLIT*D   (512 KB)

    kdot_wmma_kernel<<<dim3(Bsz, Nsz / 128), 256, 0, stream>>>(K, W, kdot);
    softmax_kernel<<<Bsz, 256, 0, stream>>>(Q, adj, smask, W, bias, rel, kdot, out);
    wsum_partial_kernel<<<dim3(Bsz, Dsz / 256, NSPLIT), 256, 0, stream>>>(V, out, part);
    wsum_reduce_kernel<<<(Bsz * Dsz) / 256, 256, 0, stream>>>(part, out);
}